// Encoder_25048249270381
// MI455X (gfx1250) — compile-verified
//
#include <hip/hip_runtime.h>

typedef __attribute__((ext_vector_type(2))) float v2f;
typedef __attribute__((ext_vector_type(8))) float v8f;

// ---------------- elementwise / degree kernels ----------------

__global__ __launch_bounds__(256) void k_fill(float* __restrict__ p, float v, int n) {
  int i = blockIdx.x * 256 + threadIdx.x;
  if (i < n) p[i] = v;
}

__global__ __launch_bounds__(256) void k_deg(const int* __restrict__ dst,
                                             float* __restrict__ deg, int E) {
  int i = blockIdx.x * 256 + threadIdx.x;
  if (i < E) atomicAdd(&deg[dst[i]], 1.0f);
}

__global__ __launch_bounds__(256) void k_rsqrt(float* __restrict__ p, int n) {
  int i = blockIdx.x * 256 + threadIdx.x;
  if (i < n) p[i] = rsqrtf(p[i]);
}

// out[i,c] = h[i,c] * dinv[i]^2   (self-loop message), C = 1<<shift channels
__global__ __launch_bounds__(256) void k_selfinit(const float* __restrict__ h,
                                                  const float* __restrict__ dinv,
                                                  float* __restrict__ out,
                                                  int n, int shift) {
  int i = blockIdx.x * 256 + threadIdx.x;
  int total = n << shift;
  if (i >= total) return;
  float w = dinv[i >> shift];
  out[i] = h[i] * w * w;
}

// buf[i,c] = act(buf[i,c] + b[c]);  C = 1<<shift, mask = C-1
__global__ __launch_bounds__(256) void k_bias_act(float* __restrict__ buf,
                                                  const float* __restrict__ b,
                                                  int n, int shift, int mask, int relu) {
  int i = blockIdx.x * 256 + threadIdx.x;
  int total = n << shift;
  if (i >= total) return;
  float v = buf[i] + b[i & mask];
  buf[i] = relu ? fmaxf(v, 0.0f) : v;
}

// ---------------- edge scatter (gather + L2 atomic add) ----------------

// 128 channels: one wave per edge, each lane owns 4 channels (float4 gather).
__global__ __launch_bounds__(256) void k_scatter128(const float* __restrict__ h,
                                                    const int* __restrict__ src,
                                                    const int* __restrict__ dst,
                                                    const float* __restrict__ dinv,
                                                    float* __restrict__ out, int E) {
  int wid = (blockIdx.x * 256 + threadIdx.x) >> 5;
  int lane = threadIdx.x & 31;
  if (wid >= E) return;
  int s = src[wid], d = dst[wid];
  float w = dinv[s] * dinv[d];
  const float4 v = *reinterpret_cast<const float4*>(h + (size_t)s * 128 + lane * 4);
  float* o = out + (size_t)d * 128 + lane * 4;
  atomicAdd(o + 0, v.x * w);
  atomicAdd(o + 1, v.y * w);
  atomicAdd(o + 2, v.z * w);
  atomicAdd(o + 3, v.w * w);
}

// 64 channels: 16 threads per edge, each owns 4 channels.
__global__ __launch_bounds__(256) void k_scatter64(const float* __restrict__ h,
                                                   const int* __restrict__ src,
                                                   const int* __restrict__ dst,
                                                   const float* __restrict__ dinv,
                                                   float* __restrict__ out, int E) {
  int gid = blockIdx.x * 256 + threadIdx.x;
  int e = gid >> 4;
  int l = gid & 15;
  if (e >= E) return;
  int s = src[e], d = dst[e];
  float w = dinv[s] * dinv[d];
  const float4 v = *reinterpret_cast<const float4*>(h + (size_t)s * 64 + l * 4);
  float* o = out + (size_t)d * 64 + l * 4;
  atomicAdd(o + 0, v.x * w);
  atomicAdd(o + 1, v.y * w);
  atomicAdd(o + 2, v.z * w);
  atomicAdd(o + 3, v.w * w);
}

// ---------------- fp32 WMMA GEMM: C[M,Nc] = A[M,K] @ B[K,Nc] (+ bias[n]) ----------------
// One wave computes one 16x16 C tile via V_WMMA_F32_16X16X4_F32, stepping K by 4.
// f32 16x4 A layout: lane L holds A[m = L&15][k0 + 2*(L>>4) + {0,1}] in 2 VGPRs.
// f32 4x16 B layout: lane L holds B[k0 + 2*(L>>4) + {0,1}][n = L&15].
// f32 16x16 C layout: VGPR r of lane L holds C[r + 8*(L>>4)][L&15].
__global__ __launch_bounds__(256) void k_gemm_wmma(const float* __restrict__ A,
                                                   const float* __restrict__ B,
                                                   const float* __restrict__ bias,
                                                   float* __restrict__ C,
                                                   int M, int K, int Nc) {
  int wid = (blockIdx.x * 256 + threadIdx.x) >> 5;
  int lane = threadIdx.x & 31;
  int nt = Nc >> 4;                 // N tiles
  int tM = wid / nt;
  int tN = wid - tM * nt;
  if (tM * 16 >= M) return;         // uniform per wave

  int r16 = lane & 15;
  int hi  = lane >> 4;              // 0 or 1

  const float* Ap = A + (size_t)(tM * 16 + r16) * K + 2 * hi;     // + k
  const float* Bp = B + (size_t)(2 * hi) * Nc + tN * 16 + r16;    // + k*Nc

  v8f acc;
  float binit = bias ? bias[tN * 16 + r16] : 0.0f;
#pragma unroll
  for (int r = 0; r < 8; ++r) acc[r] = binit;

  for (int k = 0; k < K; k += 4) {
    v2f a = *reinterpret_cast<const v2f*>(Ap + k);   // A[m][k+2hi], A[m][k+2hi+1]
    v2f b;
    b.x = Bp[(size_t)k * Nc];                        // B[k+2hi][n]
    b.y = Bp[(size_t)(k + 1) * Nc];                  // B[k+2hi+1][n]
    acc = __builtin_amdgcn_wmma_f32_16x16x4_f32(false, a, false, b,
                                                (short)0, acc, false, false);
  }

  float* Cp = C + (size_t)(tM * 16 + 8 * hi) * Nc + tN * 16 + r16;
#pragma unroll
  for (int r = 0; r < 8; ++r) Cp[(size_t)r * Nc] = acc[r];
}

// ---------------- launcher ----------------

extern "C" void kernel_launch(void* const* d_in, const int* in_sizes, int n_in,
                              void* d_out, int out_size, void* d_ws, size_t ws_size,
                              hipStream_t stream) {
  const float* x   = (const float*)d_in[0];
  const int*   ei  = (const int*)d_in[1];
  const float* W1  = (const float*)d_in[2];
  const float* b1  = (const float*)d_in[3];
  const float* W2  = (const float*)d_in[4];
  const float* b2  = (const float*)d_in[5];
  const float* Wfc = (const float*)d_in[6];
  const float* bfc = (const float*)d_in[7];
  float* out = (float*)d_out;

  const int N = in_sizes[0] / 128;   // 100000
  const int E = in_sizes[1] / 2;     // 1600000
  const int* src = ei;
  const int* dst = ei + E;

  float* dinv = (float*)d_ws;                    // N floats
  float* bufA = dinv + N;                        // N*128 floats
  float* bufB = bufA + (size_t)N * 128;          // N*128 floats

  auto cdiv = [](long long a, long long b) { return (int)((a + b - 1) / b); };

  // degrees (self-loop => start at 1.0), then dinv = rsqrt(deg)
  k_fill<<<cdiv(N, 256), 256, 0, stream>>>(dinv, 1.0f, N);
  k_deg<<<cdiv(E, 256), 256, 0, stream>>>(dst, dinv, E);
  k_rsqrt<<<cdiv(N, 256), 256, 0, stream>>>(dinv, N);

  // layer 1: h = x @ W1  (N x 128)
  {
    int waves = cdiv(N, 16) * (128 >> 4);
    k_gemm_wmma<<<cdiv(waves, 8), 256, 0, stream>>>(x, W1, nullptr, bufA, N, 128, 128);
  }
  k_selfinit<<<cdiv((long long)N << 7, 256), 256, 0, stream>>>(bufA, dinv, bufB, N, 7);
  k_scatter128<<<cdiv(E, 8), 256, 0, stream>>>(bufA, src, dst, dinv, bufB, E);
  k_bias_act<<<cdiv((long long)N << 7, 256), 256, 0, stream>>>(bufB, b1, N, 7, 127, 1);

  // layer 2: h2 = relu_h @ W2  (N x 64)
  {
    int waves = cdiv(N, 16) * (64 >> 4);
    k_gemm_wmma<<<cdiv(waves, 8), 256, 0, stream>>>(bufB, W2, nullptr, bufA, N, 128, 64);
  }
  k_selfinit<<<cdiv((long long)N << 6, 256), 256, 0, stream>>>(bufA, dinv, bufB, N, 6);
  k_scatter64<<<cdiv((long long)E * 16, 256), 256, 0, stream>>>(bufA, src, dst, dinv, bufB, E);
  k_bias_act<<<cdiv((long long)N << 6, 256), 256, 0, stream>>>(bufB, b2, N, 6, 63, 0);

  // head: out = h2 @ Wfc + bfc  (N x 64)
  {
    int waves = cdiv(N, 16) * (64 >> 4);
    k_gemm_wmma<<<cdiv(waves, 8), 256, 0, stream>>>(bufB, Wfc, bfc, out, N, 64, 64);
  }
}